// ATSSAssigner_90056874262892
// MI455X (gfx1250) — compile-verified
//
#include <hip/hip_runtime.h>
#include <stdint.h>

typedef float v2f __attribute__((ext_vector_type(2)));
typedef float v8f __attribute__((ext_vector_type(8)));

#define ATSS_TOPK 9
#define MAX_LVL 7
#define MAX_CAND (ATSS_TOPK * MAX_LVL)
#define WAVES 8
#define TPB (WAVES * 32)

__device__ __forceinline__ float iou_box(float ax0, float ay0, float ax1, float ay1,
                                         float bx0, float by0, float bx1, float by1) {
  float lx = fmaxf(ax0, bx0), ly = fmaxf(ay0, by0);
  float rx = fminf(ax1, bx1), ry = fminf(ay1, by1);
  float w = fmaxf(rx - lx, 0.0f), h = fmaxf(ry - ly, 0.0f);
  float inter = w * h;
  float aa = (ax1 - ax0) * (ay1 - ay0);
  float ab = (bx1 - bx0) * (by1 - by0);
  return inter / (aa + ab - inter + 1e-9f);
}

// lexicographic (dist, idx) top-k insertion -> order-independent, stable like lax.top_k
__device__ __forceinline__ void topk_insert(float* td, int* ti, float dv, int idx) {
  const float lw = td[ATSS_TOPK - 1];
  const int   li = ti[ATSS_TOPK - 1];
  if (dv < lw || (dv == lw && idx < li)) {
    int k = ATSS_TOPK - 1;
    while (k > 0 && (dv < td[k - 1] || (dv == td[k - 1] && idx < ti[k - 1]))) {
      td[k] = td[k - 1]; ti[k] = ti[k - 1]; --k;
    }
    td[k] = dv; ti[k] = idx;
  }
}

// Kernel 1: one workgroup (8 waves) per tile of 16 GTs.
// Each wave: async-stage anchor tiles into LDS (double-buffered, ASYNCcnt),
// WMMA 16x16 squared-distance tiles (K=4 rank trick), per-wave partial top-9,
// then LDS merge -> threshold -> atomic claim scatter.
__global__ __launch_bounds__(TPB) void atss_topk_kernel(
    const float* __restrict__ anchors,      // L x 4
    const int*   __restrict__ num_anchors,  // nlvl
    int nlvl,
    const float* __restrict__ gt_bboxes,    // (B*n) x 4
    const float* __restrict__ pad_mask,     // (B*n)
    int L, int BN, int ngt,
    int* __restrict__ count,                // B*L
    int* __restrict__ claim)                // B*L
{
  __shared__ float  distT[WAVES][16][17];
  __shared__ float4 ansh[WAVES][2][16];             // async-staged anchor tiles
  __shared__ float  mergeD[16][WAVES][ATSS_TOPK];
  __shared__ int    mergeI[16][WAVES][ATSS_TOPK];

  const int tid  = threadIdx.x;
  const int wave = tid >> 5;
  const int lane = tid & 31;
  const int m    = lane & 15;
  const int half = lane >> 4;   // 0: K pair {0,1}, 1: K pair {2,3}
  const int kk   = half * 2;
  const int gtbase = blockIdx.x * 16;

  // ---- A operand (16x4 f32): per-GT feature [gx, gy, |g|^2, 1] ----
  int giA = gtbase + m; if (giA >= BN) giA = BN - 1;
  const float4 gbA = ((const float4*)gt_bboxes)[giA];
  const float gxA = (gbA.x + gbA.z) * 0.5f;
  const float gyA = (gbA.y + gbA.w) * 0.5f;
  const float fA[4] = {gxA, gyA, gxA * gxA + gyA * gyA, 1.0f};
  v2f a; a.x = fA[kk]; a.y = fA[kk + 1];

  if (nlvl > MAX_LVL) nlvl = MAX_LVL;

  float td[ATSS_TOPK];
  int   ti[ATSS_TOPK];
  int   cand[MAX_CAND];   // meaningful in wave 0 owner lanes only
  int ncand = 0;
  int start = 0;

  for (int lvl = 0; lvl < nlvl; ++lvl) {
    const int na = num_anchors[lvl];
    if (na <= 0) continue;
    const int nchunk = (na + 15) >> 4;
    const int jmax = (nchunk + WAVES - 1) / WAVES;   // uniform across workgroup

#pragma unroll
    for (int t = 0; t < ATSS_TOPK; ++t) { td[t] = 3.0e38f; ti[t] = 0x7fffffff; }

    // ---- prologue: async-stage this wave's first chunk into buffer 0 ----
    if (lane < 16) {
      int aoff = wave * 16 + m; if (aoff >= na) aoff = na - 1;
      uint64_t ga = (uint64_t)(uintptr_t)&((const float4*)anchors)[start + aoff];
      uint32_t la = (uint32_t)(uintptr_t)&ansh[wave][0][m];
      asm volatile("global_load_async_to_lds_b128 %0, %1, off"
                   :: "v"(la), "v"(ga) : "memory");
    }

    for (int j = 0; j < jmax; ++j) {
      const int buf = j & 1;
      if (j + 1 < jmax) {
        // kick next tile into the other buffer, wait only for the current one
        if (lane < 16) {
          int aoffn = ((j + 1) * WAVES + wave) * 16 + m;
          if (aoffn >= na) aoffn = na - 1;
          uint64_t ga = (uint64_t)(uintptr_t)&((const float4*)anchors)[start + aoffn];
          uint32_t la = (uint32_t)(uintptr_t)&ansh[wave][buf ^ 1][m];
          // warm L2 one stride further ahead (global_prefetch_b8)
          int poff = aoffn + WAVES * 16; if (poff >= na) poff = na - 1;
          __builtin_prefetch((const void*)&((const float4*)anchors)[start + poff], 0, 1);
          asm volatile("global_load_async_to_lds_b128 %0, %1, off"
                       :: "v"(la), "v"(ga) : "memory");
        }
        asm volatile("s_wait_asynccnt 0x1" ::: "memory");
      } else {
        asm volatile("s_wait_asynccnt 0x0" ::: "memory");
      }

      const int c0 = (j * WAVES + wave) * 16;   // this wave's chunk (may be >= na: masked)

      // ---- B operand (4x16 f32): per-anchor feature [-2ax, -2ay, 1, |a|^2] ----
      const float4 ab = ansh[wave][buf][m];
      const float ax = (ab.x + ab.z) * 0.5f;
      const float ay = (ab.y + ab.w) * 0.5f;
      const float fB[4] = {-2.0f * ax, -2.0f * ay, 1.0f, ax * ax + ay * ay};
      v2f b; b.x = fB[kk]; b.y = fB[kk + 1];

      v8f cz = {0.f, 0.f, 0.f, 0.f, 0.f, 0.f, 0.f, 0.f};
      // D[m][n] = |g_m - a_n|^2  (gx*-2ax + gy*-2ay + |g|^2*1 + 1*|a|^2)
      v8f d2 = __builtin_amdgcn_wmma_f32_16x16x4_f32(
          false, a, false, b, (short)0, cz, false, false);

      // C layout: VGPR r -> M = r + 8*half, N = lane & 15 (per-wave private tile)
#pragma unroll
      for (int r = 0; r < 8; ++r) distT[wave][half * 8 + r][m] = d2[r];
      // intra-wave LDS is in-order; just stop the compiler from reordering
      asm volatile("" ::: "memory");

      if (lane < 16 && c0 < na) {
        int lim = na - c0; if (lim > 16) lim = 16;
        for (int q = 0; q < lim; ++q)
          topk_insert(td, ti, distT[wave][lane][q], start + c0 + q);
      }
      asm volatile("" ::: "memory");
    }

    // ---- cross-wave merge of partial top-9 lists (uniform barriers) ----
    __syncthreads();
    if (lane < 16) {
#pragma unroll
      for (int t = 0; t < ATSS_TOPK; ++t) {
        mergeD[lane][wave][t] = td[t];
        mergeI[lane][wave][t] = ti[t];
      }
    }
    __syncthreads();
    if (wave == 0 && lane < 16) {
      float fd[ATSS_TOPK]; int fi[ATSS_TOPK];
#pragma unroll
      for (int t = 0; t < ATSS_TOPK; ++t) { fd[t] = 3.0e38f; fi[t] = 0x7fffffff; }
      for (int w2 = 0; w2 < WAVES; ++w2)
        for (int t = 0; t < ATSS_TOPK; ++t)
          topk_insert(fd, fi, mergeD[lane][w2][t], mergeI[lane][w2][t]);
#pragma unroll
      for (int t = 0; t < ATSS_TOPK; ++t) cand[ncand + t] = fi[t];
    }
    __syncthreads();   // merge buffers free before next level

    ncand += ATSS_TOPK;
    start += na;
  }

  // ---- threshold + positive scatter (wave 0 owner lanes, valid GTs only) ----
  const int gidx = gtbase + lane;
  if (wave == 0 && lane < 16 && gidx < BN && pad_mask[gidx] > 0.0f) {
    const float4 gb = ((const float4*)gt_bboxes)[gidx];
    float vals[MAX_CAND];
    float s1 = 0.0f;
    for (int t = 0; t < ncand; ++t) {
      const float4 av = ((const float4*)anchors)[cand[t]];
      const float v = iou_box(gb.x, gb.y, gb.z, gb.w, av.x, av.y, av.z, av.w);
      vals[t] = v; s1 += v;
    }
    const float mean = s1 / (float)ncand;
    float s2 = 0.0f;
    for (int t = 0; t < ncand; ++t) { const float d = vals[t] - mean; s2 += d * d; }
    const float thr = mean + sqrtf(s2 / (float)(ncand - 1));  // std, ddof=1

    const int b = gidx / ngt;
    const int g_local = gidx - b * ngt;
    for (int t = 0; t < ncand; ++t) {
      if (vals[t] > thr) {
        const int ai2 = cand[t];
        const float4 av = ((const float4*)anchors)[ai2];
        const float cx = (av.x + av.z) * 0.5f;
        const float cy = (av.y + av.w) * 0.5f;
        const float dmin = fminf(fminf(cx - gb.x, cy - gb.y),
                                 fminf(gb.z - cx, gb.w - cy));
        if (dmin > 1e-9f) {   // _check_points_inside
          atomicAdd(&count[(size_t)b * L + ai2], 1);
          atomicMin(&claim[(size_t)b * L + ai2], g_local);
        }
      }
    }
  }
}

// Kernel 2: one thread per (b, anchor). Resolve claims, emit label/bbox/scores.
__global__ __launch_bounds__(256) void atss_assign_kernel(
    const float* __restrict__ anchors,
    const float* __restrict__ gt_bboxes,
    const int*   __restrict__ gt_labels,
    const float* __restrict__ pred_bboxes,
    const int*   __restrict__ bg_index_p,
    const int*   __restrict__ count,
    const int*   __restrict__ claim,
    int L, int ngt, int NC, int total,
    float* __restrict__ out_labels,
    float* __restrict__ out_bboxes,
    float* __restrict__ out_scores)
{
  const int idx = blockIdx.x * blockDim.x + threadIdx.x;
  if (idx >= total) return;
  const int b = idx / L;
  const int l = idx - b * L;

  const int c = count[idx];
  int agt = 0;
  bool pos = false;
  if (c == 1) {
    agt = claim[idx];
    pos = true;
  } else if (c > 1) {
    // mask_positive_sum > 1 -> replace with argmax IoU over ALL gts (incl. padded)
    const float4 av = ((const float4*)anchors)[l];
    float best = -1.0f;
    for (int g = 0; g < ngt; ++g) {
      const float4 gv = ((const float4*)gt_bboxes)[b * ngt + g];
      const float v = iou_box(gv.x, gv.y, gv.z, gv.w, av.x, av.y, av.z, av.w);
      if (v > best) { best = v; agt = g; }   // first-max like jnp.argmax
    }
    pos = true;
  }

  const int bg = *bg_index_p;
  const float4 gb = ((const float4*)gt_bboxes)[b * ngt + agt];
  const int lbl = pos ? gt_labels[b * ngt + agt] : bg;

  out_labels[idx] = (float)lbl;
  ((float4*)out_bboxes)[idx] = gb;   // argmax of all-zero mask -> gt 0, matches ref

  float s = 0.0f;
  if (pos) {
    const float4 pb = ((const float4*)pred_bboxes)[(size_t)b * L + l];
    s = iou_box(gb.x, gb.y, gb.z, gb.w, pb.x, pb.y, pb.z, pb.w);
  }
  // keep[] drops the bg column: class k maps to k (k<bg) or k-1 (k>bg)
  const int col = (pos && lbl != bg) ? ((lbl < bg) ? lbl : lbl - 1) : -1;
  float* srow = out_scores + (size_t)idx * NC;
  if ((NC & 3) == 0) {
    const float4 z = {0.f, 0.f, 0.f, 0.f};
    for (int k = 0; k < NC; k += 4) *(float4*)(srow + k) = z;   // b128 stores
    if (col >= 0) srow[col] = s;                                // same-wave stores in order
  } else {
    for (int k = 0; k < NC; ++k) srow[k] = (k == col) ? s : 0.0f;
  }
}

extern "C" void kernel_launch(void* const* d_in, const int* in_sizes, int n_in,
                              void* d_out, int out_size, void* d_ws, size_t ws_size,
                              hipStream_t stream) {
  const float* anchors     = (const float*)d_in[0];
  const int*   num_anchors = (const int*)  d_in[1];
  const int*   gt_labels   = (const int*)  d_in[2];
  const float* gt_bboxes   = (const float*)d_in[3];
  const float* pad_mask    = (const float*)d_in[4];
  const int*   bg_index    = (const int*)  d_in[5];
  const float* pred_bboxes = (const float*)d_in[6];

  const int L    = in_sizes[0] / 4;       // 8500
  const int nlvl = in_sizes[1];           // 4
  const int BN   = in_sizes[2];           // B*n (gt_labels is B*n*1)
  const int B    = in_sizes[6] / (4 * L); // pred_bboxes is B*L*4
  const int ngt  = BN / B;                // 64
  const int BL   = B * L;
  int NC = out_size / BL - 5;             // 80 (labels:1 + bbox:4 + scores:NC)
  if (NC < 1) NC = 1;

  int* count = (int*)d_ws;
  int* claim = count + (size_t)BL;

  hipMemsetAsync(count, 0,    (size_t)BL * sizeof(int), stream);
  hipMemsetAsync(claim, 0x7F, (size_t)BL * sizeof(int), stream);  // 0x7F7F7F7F = +inf idx

  const int blocks1 = (BN + 15) / 16;
  atss_topk_kernel<<<blocks1, TPB, 0, stream>>>(
      anchors, num_anchors, nlvl, gt_bboxes, pad_mask, L, BN, ngt, count, claim);

  float* out_labels = (float*)d_out;
  float* out_bboxes = out_labels + (size_t)BL;
  float* out_scores = out_bboxes + (size_t)BL * 4;
  atss_assign_kernel<<<(BL + 255) / 256, 256, 0, stream>>>(
      anchors, gt_bboxes, gt_labels, pred_bboxes, bg_index, count, claim,
      L, ngt, NC, BL, out_labels, out_bboxes, out_scores);
}